// GWLoss_57552561766415
// MI455X (gfx1250) — compile-verified
//
#include <hip/hip_runtime.h>

// GWLoss for MI455X (gfx1250, wave32).
// Memory-bound target: 512 MB streamed once -> ~22 us floor at 23.3 TB/s.
// One 256-thread block per row; single-pass online logsumexp carried in the
// log2 domain so each element costs 1 v_fma + 1 raw v_exp_f32 (no libm
// denorm-guard code); non-temporal b128 loads (data > 192 MB L2, touched
// once); wave32 shuffle max-reduction; WMMA f32_16x16x4 ones-matrix multiply
// as a one-shot 32-lane adder tree for the sum reductions (matrix pipe,
// co-executes with VALU).

typedef float v2f __attribute__((ext_vector_type(2)));
typedef float v4f __attribute__((ext_vector_type(4)));
typedef float v8f __attribute__((ext_vector_type(8)));

#define L2E 1.4426950408889634f   // log2(e)
#define LN2 0.6931471805599453f   // ln(2)
// 2 * variance^2 with variance = 0.1*e  ->  0.02 * e^2
#define TWO_VAR2 0.14778112197861300f

// Sum a float across all 32 lanes of a wave using the matrix pipe.
// Lane value v goes into A-matrix (16x4 f32) VGPR0: A[m][0]=v(lane m),
// A[m][2]=v(lane m+16); VGPR1 (K=1,3) = 0. B = ones(4x16).
// D[m][n] = v(lane m) + v(lane m+16). Per-lane sum of the 8 D VGPRs +
// one shfl_xor(16) yields the full 32-lane sum on every lane.
__device__ __forceinline__ float wave_sum32(float v) {
  v2f a; a[0] = v;    a[1] = 0.0f;
  v2f b; b[0] = 1.0f; b[1] = 1.0f;
  v8f c = {};
  v8f d = __builtin_amdgcn_wmma_f32_16x16x4_f32(
      /*neg_a=*/false, a, /*neg_b=*/false, b,
      /*c_mod=*/(short)0, c, /*reuse_a=*/false, /*reuse_b=*/false);
  float r = ((d[0] + d[1]) + (d[2] + d[3])) + ((d[4] + d[5]) + (d[6] + d[7]));
  r += __shfl_xor(r, 16, 32);
  return r;
}

__global__ __launch_bounds__(256)
void gw_row_kernel(const float* __restrict__ x, const int* __restrict__ tgt,
                   float* __restrict__ wout, int vocab) {
  const int row  = blockIdx.x;
  const int tid  = threadIdx.x;
  const int lane = tid & 31;
  const int wave = tid >> 5;

  const float* xr = x + (size_t)row * (size_t)vocab;
  const v4f*   xv = reinterpret_cast<const v4f*>(xr);
  const int  nvec = vocab >> 2;

  // ---- single-pass online logsumexp, carried in log2 domain ----
  // mL = running_max * log2e ; s = sum of 2^(x*log2e - mL) = sum of e^(x-max)
  float mL = -INFINITY;
  float s  = 0.0f;
  #pragma unroll 2
  for (int i = tid; i < nvec; i += 256) {
    v4f v = __builtin_nontemporal_load(&xv[i]);   // NT: streamed-once data
    float lm  = fmaxf(fmaxf(v[0], v[1]), fmaxf(v[2], v[3]));
    float lmL = lm * L2E;
    if (lmL > mL) {                                // rare after warm-up
      s *= __builtin_amdgcn_exp2f(mL - lmL);       // exp2(-inf)=0 handles init
      mL = lmL;
    }
    // raw v_exp_f32: args are <= 0 and bounded, no denorm guard needed
    float e0 = __builtin_amdgcn_exp2f(fmaf(v[0], L2E, -mL));
    float e1 = __builtin_amdgcn_exp2f(fmaf(v[1], L2E, -mL));
    float e2 = __builtin_amdgcn_exp2f(fmaf(v[2], L2E, -mL));
    float e3 = __builtin_amdgcn_exp2f(fmaf(v[3], L2E, -mL));
    s += (e0 + e1) + (e2 + e3);
  }

  // ---- wave32 max reduction (shuffles; WMMA cannot do max) ----
  float mwL = mL;
  #pragma unroll
  for (int off = 16; off >= 1; off >>= 1)
    mwL = fmaxf(mwL, __shfl_xor(mwL, off, 32));

  // rescale lane sum to wave max, then sum lanes on the matrix pipe
  float ss = s * __builtin_amdgcn_exp2f(mL - mwL);
  float sw = wave_sum32(ss);                       // EXEC all-ones here

  __shared__ float smaxL[8];
  __shared__ float ssum[8];
  if (lane == 0) { smaxL[wave] = mwL; ssum[wave] = sw; }
  __syncthreads();

  if (tid == 0) {
    float MgL = smaxL[0];
    #pragma unroll
    for (int i = 1; i < 8; ++i) MgL = fmaxf(MgL, smaxL[i]);
    float Sg = 0.0f;
    #pragma unroll
    for (int i = 0; i < 8; ++i)
      Sg += ssum[i] * __builtin_amdgcn_exp2f(smaxL[i] - MgL);
    // log2-domain logsumexp: lse2 = max*log2e + log2(Sg)
    float lse2 = MgL + __builtin_amdgcn_logf(Sg);  // v_log_f32 = log2

    int   t   = tgt[row];
    float out = 0.0f;
    if (t >= 0) {                                  // ignore_index == -1
      float xt     = xr[t];
      float logpt2 = fmaf(xt, L2E, -lse2);         // log2(p_t)
      float logpt  = logpt2 * LN2;                 // ln(p_t)
      float pt     = __builtin_amdgcn_exp2f(logpt2);
      float dd     = pt - 0.5f;
      float g      = __builtin_amdgcn_exp2f(-(dd * dd) * (L2E / TWO_VAR2));
      out = (g - 0.1f * pt) * logpt;               // gathered weighted value
    }
    wout[row] = out;
  }
}

__global__ __launch_bounds__(256)
void gw_finalize_kernel(const float* __restrict__ w, const int* __restrict__ tgt,
                        float* __restrict__ out, int n) {
  const int tid  = threadIdx.x;
  const int lane = tid & 31;
  const int wave = tid >> 5;

  float s = 0.0f, c = 0.0f;
  for (int i = tid; i < n; i += 256) {
    s += w[i];
    c += (tgt[i] != -1) ? 1.0f : 0.0f;
  }
  float sw = wave_sum32(s);                        // matrix-pipe lane sums
  float cw = wave_sum32(c);

  __shared__ float ls[8];
  __shared__ float lc[8];
  if (lane == 0) { ls[wave] = sw; lc[wave] = cw; }
  __syncthreads();

  if (tid == 0) {
    float S = 0.0f, C = 0.0f;
    #pragma unroll
    for (int i = 0; i < 8; ++i) { S += ls[i]; C += lc[i]; }
    out[0] = (C > 0.0f) ? (-S / C) : 0.0f;         // mean over valid rows
  }
}

extern "C" void kernel_launch(void* const* d_in, const int* in_sizes, int n_in,
                              void* d_out, int out_size, void* d_ws, size_t ws_size,
                              hipStream_t stream) {
  const float* input  = (const float*)d_in[0];
  const int*   target = (const int*)d_in[1];
  const int n     = in_sizes[1];                   // 4096 rows
  const int vocab = in_sizes[0] / n;               // 32000 classes

  float* wrow = (float*)d_ws;                      // per-row gathered value

  gw_row_kernel<<<n, 256, 0, stream>>>(input, target, wrow, vocab);
  gw_finalize_kernel<<<1, 256, 0, stream>>>(wrow, target, (float*)d_out, n);
}